// SegmentedPolynomial_46497315947084
// MI455X (gfx1250) — compile-verified
//
#include <hip/hip_runtime.h>

typedef __attribute__((ext_vector_type(2))) float v2f;
typedef __attribute__((ext_vector_type(4))) float v4f;
typedef __attribute__((ext_vector_type(8))) float v8f;

#define BATCH        16384
#define NUM_ELEMENTS 64
#define IN_SIZE      448
#define W_SIZE       14336

// Workspace layout (ints): [0,64) counts | [64,128) offsets | [128,192) cursors | [192,192+BATCH) order

__global__ void k_hist(const int* __restrict__ idx, int* __restrict__ counts) {
    int b = blockIdx.x * blockDim.x + threadIdx.x;
    if (b < BATCH) atomicAdd(&counts[idx[b] & 63], 1);
}

__global__ void k_scan(const int* __restrict__ counts, int* __restrict__ offsets) {
    if (threadIdx.x == 0) {
        int s = 0;
        for (int e = 0; e < NUM_ELEMENTS; ++e) { offsets[e] = s; s += counts[e]; }
    }
}

__global__ void k_scatter(const int* __restrict__ idx, const int* __restrict__ offsets,
                          int* __restrict__ cursors, int* __restrict__ order) {
    int b = blockIdx.x * blockDim.x + threadIdx.x;
    if (b < BATCH) {
        int e = idx[b] & 63;
        int p = atomicAdd(&cursors[e], 1);
        order[offsets[e] + p] = b;
    }
}

// One (element e, o-tile mt) per block for one segment. A-fragments (weights)
// are preloaded into registers once; 8 waves stride over N-tiles of the
// bucket (N = flattened (node, m) columns).
// Inner GEMM: D(16x16) += A(16x4 = W^T tile) * B(4x16 = x tile), f32 WMMA.
template<int MI, int MO, int D, int WSEG, int XOFF>
__device__ __forceinline__ void gemm_seg(int e, int mt,
                                         const float* __restrict__ W,
                                         const float* __restrict__ X,
                                         const int* __restrict__ counts,
                                         const int* __restrict__ offsets,
                                         const int* __restrict__ order,
                                         float* __restrict__ Y) {
    constexpr int KSTEPS = MI / 4;

    const int lane = threadIdx.x & 31;
    const int wave = threadIdx.x >> 5;   // 0..7
    const int half = lane >> 4;          // K-pair select (A/B); M-half select (D)
    const int l16  = lane & 15;

    const int cnt    = counts[e];
    const int base   = offsets[e];
    const int ncols  = cnt * D;
    const int ntiles = (ncols + 15) >> 4;
    const int o0     = mt << 4;

    // ---- Preload A fragments (invariant over N-tiles). Constant offsets. ----
    const float* wbase = W + e * W_SIZE + WSEG + o0 + l16;
    v2f a[KSTEPS];
    #pragma unroll
    for (int kk = 0; kk < KSTEPS; ++kk) {
        a[kk].x = wbase[(4 * kk + 0) * MO + half * 2 * MO];
        a[kk].y = wbase[(4 * kk + 1) * MO + half * 2 * MO];
    }

    for (int nt = wave; nt < ntiles; nt += 8) {
        const int n0  = nt << 4;
        const int col = n0 + l16;
        v8f c = {0.f, 0.f, 0.f, 0.f, 0.f, 0.f, 0.f, 0.f};

        if (n0 + 16 <= ncols) {
            // -------- full tile: no clamping, no masking --------
            const int nidx = col / D;
            const int m    = col - nidx * D;
            const int node = order[base + nidx];
            const float* xrow = X + node * IN_SIZE + XOFF + m;

            #pragma unroll
            for (int kk = 0; kk < KSTEPS; ++kk) {
                v2f b;
                if constexpr (D == 1) {
                    // contiguous K pair -> single b64 load (8B aligned)
                    b = *(const v2f*)(xrow + 4 * kk + half * 2);
                } else {
                    b.x = xrow[(4 * kk + half * 2 + 0) * D];
                    b.y = xrow[(4 * kk + half * 2 + 1) * D];
                }
                c = __builtin_amdgcn_wmma_f32_16x16x4_f32(
                        false, a[kk], false, b, (short)0, c, false, false);
            }

            float* yrow = Y + node * IN_SIZE + XOFF + m;
            if constexpr (D == 1) {
                // 8 contiguous outputs per lane -> two b128 stores (16B aligned)
                v4f lo = {c[0], c[1], c[2], c[3]};
                v4f hi = {c[4], c[5], c[6], c[7]};
                *(v4f*)(yrow + o0 + half * 8 + 0) = lo;
                *(v4f*)(yrow + o0 + half * 8 + 4) = hi;
            } else {
                #pragma unroll
                for (int j = 0; j < 8; ++j)
                    yrow[(o0 + half * 8 + j) * D] = c[j];
            }
        } else {
            // -------- tail tile: clamp addresses, mask B, predicated stores --------
            const bool valid = col < ncols;
            const int  colc  = valid ? col : (ncols - 1);
            const int  nidx  = colc / D;
            const int  m     = colc - nidx * D;
            const int  node  = order[base + nidx];
            const float* xrow = X + node * IN_SIZE + XOFF + m;

            #pragma unroll
            for (int kk = 0; kk < KSTEPS; ++kk) {
                v2f b;
                if constexpr (D == 1) {
                    v2f t = *(const v2f*)(xrow + 4 * kk + half * 2);
                    b.x = valid ? t.x : 0.f;
                    b.y = valid ? t.y : 0.f;
                } else {
                    const float bx = xrow[(4 * kk + half * 2 + 0) * D];
                    const float by = xrow[(4 * kk + half * 2 + 1) * D];
                    b.x = valid ? bx : 0.f;
                    b.y = valid ? by : 0.f;
                }
                c = __builtin_amdgcn_wmma_f32_16x16x4_f32(
                        false, a[kk], false, b, (short)0, c, false, false);
            }

            if (valid) {
                float* yrow = Y + node * IN_SIZE + XOFF + m;
                #pragma unroll
                for (int j = 0; j < 8; ++j)
                    yrow[(o0 + half * 8 + j) * D] = c[j];
            }
        }
    }
}

// Grid: 384 blocks seg0 (64e x 6 o-tiles) | 256 seg1 (64e x 4) | 128 seg2 (64e x 2)
__global__ void __launch_bounds__(256)
k_gemm(const float* __restrict__ W, const float* __restrict__ X,
       const int* __restrict__ counts, const int* __restrict__ offsets,
       const int* __restrict__ order, float* __restrict__ Y) {
    const int bid = blockIdx.x;
    if (bid < 384) {
        gemm_seg<96, 96, 1, 0, 0>(bid / 6, bid % 6, W, X, counts, offsets, order, Y);
    } else if (bid < 640) {
        const int r = bid - 384;
        gemm_seg<64, 64, 3, 9216, 96>(r / 4, r % 4, W, X, counts, offsets, order, Y);
    } else {
        const int r = bid - 640;
        gemm_seg<32, 32, 5, 13312, 288>(r / 2, r % 2, W, X, counts, offsets, order, Y);
    }
}

extern "C" void kernel_launch(void* const* d_in, const int* in_sizes, int n_in,
                              void* d_out, int out_size, void* d_ws, size_t ws_size,
                              hipStream_t stream) {
    const float* W   = (const float*)d_in[0];
    const float* X   = (const float*)d_in[1];
    const int*   idx = (const int*)d_in[2];
    float*       Y   = (float*)d_out;

    int* ws      = (int*)d_ws;
    int* counts  = ws;
    int* offsets = ws + 64;
    int* cursors = ws + 128;
    int* order   = ws + 192;

    hipMemsetAsync(ws, 0, 192 * sizeof(int), stream);
    k_hist   <<<BATCH / 256, 256, 0, stream>>>(idx, counts);
    k_scan   <<<1, 32, 0, stream>>>(counts, offsets);
    k_scatter<<<BATCH / 256, 256, 0, stream>>>(idx, offsets, cursors, order);
    k_gemm   <<<384 + 256 + 128, 256, 0, stream>>>(W, X, counts, offsets, order, Y);
}